// patchInstanceNormfbCNN_11905649344664
// MI455X (gfx1250) — compile-verified
//
#include <hip/hip_runtime.h>
#include <math.h>

typedef __attribute__((ext_vector_type(2))) float v2f;
typedef __attribute__((ext_vector_type(8))) float v8f;

#define EPSF 1e-5f

#if defined(__has_builtin)
#if __has_builtin(__builtin_amdgcn_global_load_async_to_lds_b32)
#define HAVE_ASYNC_LDS 1
#else
#define HAVE_ASYNC_LDS 0
#endif
#else
#define HAVE_ASYNC_LDS 0
#endif

typedef __attribute__((address_space(1))) int as1_int;
typedef __attribute__((address_space(3))) int as3_int;

__device__ __forceinline__ void wait_async_cnt0() {
#if defined(__has_builtin)
#if __has_builtin(__builtin_amdgcn_s_wait_asynccnt)
  __builtin_amdgcn_s_wait_asynccnt(0);
  return;
#endif
#endif
#if HAVE_ASYNC_LDS
  asm volatile("s_wait_asynccnt 0" ::: "memory");
#endif
}

// ---------------------------------------------------------------------------
// Implicit-GEMM convolution on V_WMMA_F32_16X16X4_F32.
// Rows M = flattened (b, oy, ox); K = IC*KH*KW; N = OC (padded to 16).
// One wave = one 16-row M tile, all N tiles.
//  - Weights staged into LDS pair-interleaved (B layout), via async-to-LDS
//    when available (ASYNCcnt path), so each lane reads its (kb,kb+1) B pair
//    with one ds_load_b64.
//  - Per-k im2col decode precomputed into an LDS table so the inner loop is
//    one ds_load_b128 + 2 bound checks + 2 gathers per K slab (no div/mod).
// All grid sizes divide exactly -> EXEC all ones at every WMMA.
// ---------------------------------------------------------------------------
template <int IC, int OC, int KH, int KW, int IH, int IW, int OH, int OW,
          int STR, int PAD>
__global__ __launch_bounds__(256) void conv_wmma_kernel(
    const float* __restrict__ in, const float* __restrict__ w,
    const float* __restrict__ bias, float* __restrict__ out) {
  constexpr int K      = IC * KH * KW;
  constexpr int KSTEPS = (K + 3) / 4;
  constexpr int KPADD  = KSTEPS * 4;
  constexpr int OCP    = (OC < 16) ? 16 : OC;
  constexpr int NT     = OCP / 16;

  // pair-interleaved B: ldsw[((k>>1)*OCP + oc)*2 + (k&1)]
  __shared__ __align__(16) float ldsw[KPADD * OCP];
  // per-k gather table: tbl[2k] = (ic*IH+ky)*IW+kx ; tbl[2k+1] = (ky<<16)|kx
  __shared__ __align__(16) int tbl[KPADD * 2];

  const int tid = threadIdx.x;

  for (int k = tid; k < KPADD; k += 256) {
    int off  = 0;
    int code = (0x7fff << 16) | 0x7fff;  // sentinel: always out of bounds
    if (k < K) {
      const int ic = k / (KH * KW);
      const int rr = k - ic * (KH * KW);
      const int ky = rr / KW;
      const int kx = rr - ky * KW;
      off  = (ic * IH + ky) * IW + kx;
      code = (ky << 16) | kx;
    }
    tbl[2 * k]     = off;
    tbl[2 * k + 1] = code;
  }

  for (int idx = tid; idx < KPADD * OCP; idx += 256) {
    const int k  = idx / OCP;
    const int oc = idx - k * OCP;
    const int di = ((k >> 1) * OCP + oc) * 2 + (k & 1);
    if (k < K && oc < OC) {
#if HAVE_ASYNC_LDS
      __builtin_amdgcn_global_load_async_to_lds_b32(
          (as1_int*)(void*)(w + oc * K + k),
          (as3_int*)(void*)&ldsw[di], 0, 0);
#else
      ldsw[di] = w[oc * K + k];
#endif
    } else {
      ldsw[di] = 0.0f;
    }
  }
  wait_async_cnt0();
  __syncthreads();

  const int wave  = tid >> 5;
  const int lane  = tid & 31;
  const int mtile = blockIdx.x * 8 + wave;
  const int m     = mtile * 16 + (lane & 15);

  // decode row -> (b, oy, ox); constant across the K loop
  const int b   = m / (OH * OW);
  const int r0  = m - b * (OH * OW);
  const int oy  = r0 / OW;
  const int ox  = r0 - oy * OW;
  const int ihb = oy * STR - PAD;
  const int iwb = ox * STR - PAD;
  const int koff    = (lane >> 4) << 1;  // lanes 16-31 carry K=2,3 of each slab
  const int laneoff = ihb * IW + iwb;
  const float* __restrict__ inb = in + (size_t)b * IC * IH * IW;

  v8f acc[NT] = {};
  for (int s = 0; s < KSTEPS; ++s) {
    const int kb = s * 4 + koff;  // even
    const int4 te = *(const int4*)&tbl[2 * kb];
    v2f a;
    {
      const int iy = ihb + (te.y >> 16);
      const int ix = iwb + (te.y & 0xffff);
      a.x = ((unsigned)iy < (unsigned)IH && (unsigned)ix < (unsigned)IW)
                ? inb[laneoff + te.x] : 0.0f;
    }
    {
      const int iy = ihb + (te.w >> 16);
      const int ix = iwb + (te.w & 0xffff);
      a.y = ((unsigned)iy < (unsigned)IH && (unsigned)ix < (unsigned)IW)
                ? inb[laneoff + te.z] : 0.0f;
    }
#pragma unroll
    for (int t = 0; t < NT; ++t) {
      const int n  = t * 16 + (lane & 15);
      const v2f bm = *(const v2f*)&ldsw[((kb >> 1) * OCP + n) * 2];
      acc[t] = __builtin_amdgcn_wmma_f32_16x16x4_f32(
          false, a, false, bm, (short)0, acc[t], false, false);
    }
  }

  // C/D layout: VGPR r -> M = r (+8 for lanes 16-31), N = lane&15
#pragma unroll
  for (int t = 0; t < NT; ++t) {
    const int n = t * 16 + (lane & 15);
    if (n < OC) {
      const float bs = bias[n];
#pragma unroll
      for (int r = 0; r < 8; ++r) {
        const int row = mtile * 16 + r + ((lane & 16) ? 8 : 0);
        const int bb  = row / (OH * OW);
        const int rr  = row - bb * (OH * OW);
        const int yy  = rr / OW;
        const int xx  = rr - yy * OW;
        out[(((size_t)bb * OC + n) * OH + yy) * OW + xx] = acc[t][r] + bs;
      }
    }
  }
}

// ---------------------------------------------------------------------------
// PIN 0: normalize x0 [256,8,56,56]. first: own 5x5 window stats (52x52 map,
// nearest-upsampled). else: meanH/varH from mc1/vc1 [256,16,18,18], spatial
// nearest 18->56, channel stride-2 slice (target c -> source 2c).
// ---------------------------------------------------------------------------
__global__ __launch_bounds__(256) void pin0_kernel(
    const float* __restrict__ x0, const float* __restrict__ g0,
    const float* __restrict__ b0, const float* __restrict__ mc1,
    const float* __restrict__ vc1, int first, float* __restrict__ h0) {
  const int idx = blockIdx.x * 256 + threadIdx.x;  // exactly 256*8*56*56
  int t = idx;
  const int x = t % 56; t /= 56;
  const int y = t % 56; t /= 56;
  const int c = t % 8;
  const int b = t / 8;
  const float xv = x0[idx];
  float mean, var;
  if (first) {
    const int ry = (y * 52) / 56;
    const int cx = (x * 52) / 56;
    const float* base = x0 + (((size_t)b * 8 + c) * 56 + ry) * 56 + cx;
    float s1 = 0.f, s2 = 0.f;
#pragma unroll
    for (int dy = 0; dy < 5; ++dy)
#pragma unroll
      for (int dx = 0; dx < 5; ++dx) {
        const float v = base[dy * 56 + dx];
        s1 += v; s2 += v * v;
      }
    mean = s1 * (1.0f / 25.0f);
    var  = (s2 - s1 * s1 * (1.0f / 25.0f)) * (1.0f / 24.0f);
  } else {
    const int ty = (y * 18) / 56;
    const int tx = (x * 18) / 56;
    const int si = (((b * 16) + 2 * c) * 18 + ty) * 18 + tx;
    mean = mc1[si];
    var  = vc1[si];
  }
  const int gi = (c * 56 + y) * 56 + x;
  const float val = (xv - mean) * rsqrtf(var + EPSF) * g0[gi] + b0[gi];
  h0[idx] = fmaxf(val, 0.0f);
}

// ---------------------------------------------------------------------------
// PIN 1: c1 [256,16,18,18] -> mc1, vc1 (own stats, 14x14 map upsampled) and
// h1 normalized with meanH/varH (first: own stats; else from mc2/vc2
// [256,32,18,18], channel 2c, identity spatial).
// ---------------------------------------------------------------------------
__global__ __launch_bounds__(256) void pin1_kernel(
    const float* __restrict__ c1, const float* __restrict__ g1,
    const float* __restrict__ b1, const float* __restrict__ mc2,
    const float* __restrict__ vc2, int first, float* __restrict__ mc1,
    float* __restrict__ vc1, float* __restrict__ h1) {
  const int idx = blockIdx.x * 256 + threadIdx.x;  // 256*16*18*18
  int t = idx;
  const int x = t % 18; t /= 18;
  const int y = t % 18; t /= 18;
  const int c = t % 16;
  const int b = t / 16;
  const float cv = c1[idx];
  const int ry = (y * 14) / 18;
  const int cx = (x * 14) / 18;
  const float* base = c1 + (((size_t)b * 16 + c) * 18 + ry) * 18 + cx;
  float s1 = 0.f, s2 = 0.f;
#pragma unroll
  for (int dy = 0; dy < 5; ++dy)
#pragma unroll
    for (int dx = 0; dx < 5; ++dx) {
      const float v = base[dy * 18 + dx];
      s1 += v; s2 += v * v;
    }
  const float mean = s1 * (1.0f / 25.0f);
  const float var  = (s2 - s1 * s1 * (1.0f / 25.0f)) * (1.0f / 24.0f);
  mc1[idx] = mean;
  vc1[idx] = var;
  float mH, vH;
  if (first) {
    mH = mean; vH = var;
  } else {
    const int si = (((b * 32) + 2 * c) * 18 + y) * 18 + x;
    mH = mc2[si]; vH = vc2[si];
  }
  const int gi = (c * 18 + y) * 18 + x;
  h1[idx] = fmaxf((cv - mH) * rsqrtf(vH + EPSF) * g1[gi] + b1[gi], 0.0f);
}

// ---------------------------------------------------------------------------
// PIN 2: c2 [256,32,18,18] -> mc2, vc2 and h2 (always own stats).
// ---------------------------------------------------------------------------
__global__ __launch_bounds__(256) void pin2_kernel(
    const float* __restrict__ c2, const float* __restrict__ g2,
    const float* __restrict__ b2, float* __restrict__ mc2,
    float* __restrict__ vc2, float* __restrict__ h2) {
  const int idx = blockIdx.x * 256 + threadIdx.x;  // 256*32*18*18
  int t = idx;
  const int x = t % 18; t /= 18;
  const int y = t % 18; t /= 18;
  const int c = t % 32;
  const int b = t / 32;
  const float cv = c2[idx];
  const int ry = (y * 14) / 18;
  const int cx = (x * 14) / 18;
  const float* base = c2 + (((size_t)b * 32 + c) * 18 + ry) * 18 + cx;
  float s1 = 0.f, s2 = 0.f;
#pragma unroll
  for (int dy = 0; dy < 5; ++dy)
#pragma unroll
    for (int dx = 0; dx < 5; ++dx) {
      const float v = base[dy * 18 + dx];
      s1 += v; s2 += v * v;
    }
  const float mean = s1 * (1.0f / 25.0f);
  const float var  = (s2 - s1 * s1 * (1.0f / 25.0f)) * (1.0f / 24.0f);
  mc2[idx] = mean;
  vc2[idx] = var;
  const int gi = (c * 18 + y) * 18 + x;
  h2[idx] = fmaxf((cv - mean) * rsqrtf(var + EPSF) * g2[gi] + b2[gi], 0.0f);
}

// ---------------------------------------------------------------------------
// BatchNorm-over-batch stats folded into per-feature scale/offset.
// ---------------------------------------------------------------------------
__global__ __launch_bounds__(256) void bnstats_kernel(
    const float* __restrict__ h2, const float* __restrict__ bn_w,
    const float* __restrict__ bn_b, float* __restrict__ scale,
    float* __restrict__ off) {
  const int j = blockIdx.x * 256 + threadIdx.x;
  if (j < 10368) {
    float s1 = 0.f, s2 = 0.f;
    for (int b = 0; b < 256; ++b) {
      const float v = h2[(size_t)b * 10368 + j];
      s1 += v; s2 += v * v;
    }
    const float m   = s1 * (1.0f / 256.0f);
    const float var = s2 * (1.0f / 256.0f) - m * m;  // biased
    const float sc  = bn_w[j] * rsqrtf(var + EPSF);
    scale[j] = sc;
    off[j]   = bn_b[j] - m * sc;
  }
}

__global__ __launch_bounds__(256) void foldw_kernel(
    const float* __restrict__ fc_w, const float* __restrict__ scale,
    float* __restrict__ wp) {
  const int idx = blockIdx.x * 256 + threadIdx.x;  // 32*10368
  const int j = idx % 10368;
  wp[idx] = fc_w[idx] * scale[j];
}

__global__ __launch_bounds__(256) void bprime_kernel(
    const float* __restrict__ fc_w, const float* __restrict__ fc_b,
    const float* __restrict__ off, float* __restrict__ bp) {
  __shared__ float red[256];
  const int k = blockIdx.x;  // 0..31
  float s = 0.f;
  for (int j = threadIdx.x; j < 10368; j += 256)
    s += off[j] * fc_w[(size_t)k * 10368 + j];
  red[threadIdx.x] = s;
  __syncthreads();
  for (int d = 128; d > 0; d >>= 1) {
    if (threadIdx.x < d) red[threadIdx.x] += red[threadIdx.x + d];
    __syncthreads();
  }
  if (threadIdx.x == 0) bp[k] = fc_b[k] + red[0];
}

__global__ __launch_bounds__(256) void preinit_kernel(
    const float* __restrict__ bp, float* __restrict__ pre) {
  const int idx = blockIdx.x * 256 + threadIdx.x;  // 256*32
  pre[idx] = bp[idx & 31];
}

// ---------------------------------------------------------------------------
// FC GEMM: [256,10368] x [10368,32] via WMMA f32. 16 blocks = 16 M tiles;
// 8 waves split K (324 ksteps each), accumulate with global f32 atomics.
// kb is even -> A/B pair fetches are single 8B-aligned b64 loads.
// ---------------------------------------------------------------------------
__global__ __launch_bounds__(256) void fc_wmma_kernel(
    const float* __restrict__ A, const float* __restrict__ Wp,
    float* __restrict__ pre) {
  const int tid   = threadIdx.x;
  const int wave  = tid >> 5;
  const int lane  = tid & 31;
  const int mtile = blockIdx.x;  // 0..15
  const int m     = mtile * 16 + (lane & 15);
  const int koff  = (lane >> 4) << 1;
  const int sbeg  = wave * 324;  // 2592 ksteps / 8 waves
  v8f acc[2] = {};
  for (int s = 0; s < 324; ++s) {
    const int kb = (sbeg + s) * 4 + koff;
    const v2f a  = *(const v2f*)(A + (size_t)m * 10368 + kb);
#pragma unroll
    for (int t = 0; t < 2; ++t) {
      const int n  = t * 16 + (lane & 15);
      const v2f bm = *(const v2f*)(Wp + (size_t)n * 10368 + kb);
      acc[t] = __builtin_amdgcn_wmma_f32_16x16x4_f32(
          false, a, false, bm, (short)0, acc[t], false, false);
    }
  }
#pragma unroll
  for (int t = 0; t < 2; ++t) {
    const int n = t * 16 + (lane & 15);
#pragma unroll
    for (int r = 0; r < 8; ++r) {
      const int row = mtile * 16 + r + ((lane & 16) ? 8 : 0);
      atomicAdd(&pre[row * 32 + n], acc[t][r]);
    }
  }
}

// ---------------------------------------------------------------------------
// relu + two small heads + softmax. One thread per batch row.
// ---------------------------------------------------------------------------
__global__ __launch_bounds__(256) void heads_kernel(
    const float* __restrict__ pre, const float* __restrict__ shw,
    const float* __restrict__ shb, const float* __restrict__ vw,
    const float* __restrict__ vb, float* __restrict__ out) {
  const int b = threadIdx.x;  // 256
  float hid[32];
#pragma unroll
  for (int k = 0; k < 32; ++k) hid[k] = fmaxf(pre[b * 32 + k], 0.0f);

  float lg[5];
  float mx = -3.4e38f;
#pragma unroll
  for (int s = 0; s < 5; ++s) {
    float l = shb[s];
#pragma unroll
    for (int k = 0; k < 32; ++k) l += hid[k] * shw[s * 32 + k];
    lg[s] = l;
    mx = fmaxf(mx, l);
  }
  float den = 0.f;
#pragma unroll
  for (int s = 0; s < 5; ++s) { lg[s] = expf(lg[s] - mx); den += lg[s]; }
#pragma unroll
  for (int s = 0; s < 5; ++s) out[b * 5 + s] = lg[s] / den;

  float vg[2];
  float mv = -3.4e38f;
#pragma unroll
  for (int s = 0; s < 2; ++s) {
    float l = vb[s];
#pragma unroll
    for (int k = 0; k < 32; ++k) l += hid[k] * vw[s * 32 + k];
    vg[s] = l;
    mv = fmaxf(mv, l);
  }
  float dv = 0.f;
#pragma unroll
  for (int s = 0; s < 2; ++s) { vg[s] = expf(vg[s] - mv); dv += vg[s]; }
#pragma unroll
  for (int s = 0; s < 2; ++s) out[1280 + b * 2 + s] = vg[s] / dv;
}

// ---------------------------------------------------------------------------
extern "C" void kernel_launch(void* const* d_in, const int* in_sizes, int n_in,
                              void* d_out, int out_size, void* d_ws,
                              size_t ws_size, hipStream_t stream) {
  const float* x    = (const float*)d_in[0];
  const float* c0w  = (const float*)d_in[1];
  const float* c0b  = (const float*)d_in[2];
  const float* c1w  = (const float*)d_in[3];
  const float* c1b  = (const float*)d_in[4];
  const float* c2w  = (const float*)d_in[5];
  const float* c2b  = (const float*)d_in[6];
  const float* g0   = (const float*)d_in[7];
  const float* b0   = (const float*)d_in[8];
  const float* g1   = (const float*)d_in[9];
  const float* b1   = (const float*)d_in[10];
  const float* g2   = (const float*)d_in[11];
  const float* b2   = (const float*)d_in[12];
  const float* bn_w = (const float*)d_in[13];
  const float* bn_b = (const float*)d_in[14];
  const float* fc_w = (const float*)d_in[15];
  const float* fc_b = (const float*)d_in[16];
  const float* shw  = (const float*)d_in[17];
  const float* shb  = (const float*)d_in[18];
  const float* vw   = (const float*)d_in[19];
  const float* vb   = (const float*)d_in[20];
  float* out = (float*)d_out;
  float* ws  = (float*)d_ws;

  constexpr size_t N_X0 = (size_t)256 * 8 * 56 * 56;   // 6422528
  constexpr size_t N_C1 = (size_t)256 * 16 * 18 * 18;  // 1327104
  constexpr size_t N_C2 = (size_t)256 * 32 * 18 * 18;  // 2654208

  float* x0    = ws;                 // conv0 out
  float* h0    = x0 + N_X0;          // pin0 out
  float* c1    = h0 + N_X0;          // conv1 out
  float* mc1   = c1 + N_C1;
  float* vc1   = mc1 + N_C1;
  float* h1    = vc1 + N_C1;         // pin1 out
  float* c2    = h1 + N_C1;          // conv2 out
  float* mc2   = c2 + N_C2;
  float* vc2   = mc2 + N_C2;
  float* h2    = vc2 + N_C2;         // pin2 out == flat [256,10368]
  float* scale = h2 + N_C2;          // 10368
  float* offb  = scale + 10368;      // 10368
  float* wp    = offb + 10368;       // 32*10368
  float* bp    = wp + (size_t)32 * 10368;  // 32
  float* pre   = bp + 32;            // 256*32

  // conv0: rows = 256*56*56 = 802816 -> 50176 M tiles -> 6272 blocks
  conv_wmma_kernel<3, 8, 7, 7, 224, 224, 56, 56, 4, 2>
      <<<6272, 256, 0, stream>>>(x, c0w, c0b, x0);

  for (int it = 0; it < 4; ++it) {
    const int first = (it == 0) ? 1 : 0;
    pin0_kernel<<<25088, 256, 0, stream>>>(x0, g0, b0, mc1, vc1, first, h0);
    // conv1: rows = 256*18*18 = 82944 -> 5184 M tiles -> 648 blocks
    conv_wmma_kernel<8, 16, 5, 5, 56, 56, 18, 18, 3, 1>
        <<<648, 256, 0, stream>>>(h0, c1w, c1b, c1);
    pin1_kernel<<<5184, 256, 0, stream>>>(c1, g1, b1, mc2, vc2, first, mc1,
                                          vc1, h1);
    conv_wmma_kernel<16, 32, 3, 3, 18, 18, 18, 18, 1, 1>
        <<<648, 256, 0, stream>>>(h1, c2w, c2b, c2);
    pin2_kernel<<<10368, 256, 0, stream>>>(c2, g2, b2, mc2, vc2, h2);
  }

  bnstats_kernel<<<41, 256, 0, stream>>>(h2, bn_w, bn_b, scale, offb);
  foldw_kernel<<<1296, 256, 0, stream>>>(fc_w, scale, wp);
  bprime_kernel<<<32, 256, 0, stream>>>(fc_w, fc_b, offb, bp);
  preinit_kernel<<<32, 256, 0, stream>>>(bp, pre);
  fc_wmma_kernel<<<16, 256, 0, stream>>>(h2, wp, pre);
  heads_kernel<<<1, 256, 0, stream>>>(pre, shw, shb, vw, vb, out);
}